// ModelNew_25056839205222
// MI455X (gfx1250) — compile-verified
//
#include <hip/hip_runtime.h>

typedef __attribute__((ext_vector_type(2))) float v2f;
typedef __attribute__((ext_vector_type(8))) float v8f;

// Shapes: o[N=16, C=32, D=32, H=64, W=64]  bias[C=32]
// Slab = one (n, c, pd): two D-planes of 64x64 fp32 = 32 KB, 8192 slabs total.
#define N_B        16
#define C_B        32
#define PD_B       16
#define NSLABS     (N_B * C_B * PD_B)        // 8192
#define PARTS_PER_N (C_B * PD_B)             // 512
#define SLAB_FLOATS 8192                     // 2 * 64 * 64

// ---------------------------------------------------------------------------
// Kernel 1: bandwidth kernel. One block per (n,c,pd) slab. Reads the slab
// exactly once with float4 (global_load_b128) coalesced loads, computes the
// 2x2x2 max-pool and the per-slab sum of pooled maxes. Deterministic
// reduction (shfl within wave32, LDS across waves), no atomics.
// ---------------------------------------------------------------------------
__global__ __launch_bounds__(256) void pool_partial_kernel(
    const float* __restrict__ o, float* __restrict__ partials) {
  const int bid = blockIdx.x;  // n-major: bid = (n*32 + c)*16 + pd
  const float4* __restrict__ p =
      (const float4*)(o + (size_t)bid * SLAB_FLOATS);

  // Plane of 64x64 floats = 1024 float4; row = 16 float4.
  float s = 0.0f;
#pragma unroll
  for (int it = 0; it < 2; ++it) {
    const int u  = threadIdx.x + it * 256;  // 0..511 pooled float4-units
    const int ph = u >> 4;                  // pooled row 0..31
    const int u4 = u & 15;                  // float4 column (2 pooled outputs)
    const int r  = (ph << 5) + u4;          // (2*ph)*16 + u4

    float4 a = p[r];            // d=0, h=2ph
    float4 b = p[r + 16];       // d=0, h=2ph+1
    float4 c = p[r + 1024];     // d=1, h=2ph
    float4 d = p[r + 1040];     // d=1, h=2ph+1

    float mx = fmaxf(fmaxf(a.x, b.x), fmaxf(c.x, d.x));
    float my = fmaxf(fmaxf(a.y, b.y), fmaxf(c.y, d.y));
    float mz = fmaxf(fmaxf(a.z, b.z), fmaxf(c.z, d.z));
    float mw = fmaxf(fmaxf(a.w, b.w), fmaxf(c.w, d.w));
    s += fmaxf(mx, my) + fmaxf(mz, mw);
  }

  // wave32 reduction
#pragma unroll
  for (int off = 16; off > 0; off >>= 1) s += __shfl_xor(s, off, 32);

  __shared__ float wsum[8];
  const int lane = threadIdx.x & 31;
  const int wid  = threadIdx.x >> 5;
  if (lane == 0) wsum[wid] = s;
  __syncthreads();
  if (threadIdx.x == 0) {
    float t = 0.0f;
#pragma unroll
    for (int i = 0; i < 8; ++i) t += wsum[i];
    partials[bid] = t;  // raw sum of maxes (no /2 yet)
  }
}

// ---------------------------------------------------------------------------
// Kernel 2: finalize with V_WMMA_F32_16X16X4_F32 (fp32, no precision loss).
// partials is a 16x512 matrix (row = batch n). Multiplying by an all-ones
// B (4x16) and accumulating over 128 K=4 steps puts each row's total sum in
// every column of the 16x16 f32 accumulator.
// A layout (ISA 7.12.2, 32-bit A 16x4): lanes 0-15 -> M=lane, VGPR0=K0,
// VGPR1=K1; lanes 16-31 -> M=lane-16, K2/K3.
// C/D layout: lanes 0-15 = column N; VGPR v -> row M=v (lanes 0-15),
// M=v+8 (lanes 16-31). So lane 0 holds rows 0-7, lane 16 holds rows 8-15
// (column 0). One wave, EXEC all ones as WMMA requires.
// ---------------------------------------------------------------------------
__global__ __launch_bounds__(32) void finalize_wmma_kernel(
    const float* __restrict__ partials, const float* __restrict__ bias,
    float* __restrict__ out) {
  const int lane = threadIdx.x;            // 0..31
  const int m    = lane & 15;              // matrix row handled by this lane
  const int k0   = (lane < 16) ? 0 : 2;    // K pair within each K=4 step

  const float* __restrict__ row = partials + (size_t)m * PARTS_PER_N;

  v8f acc = {};
  v2f ones;
  ones.x = 1.0f;
  ones.y = 1.0f;

#pragma unroll 4
  for (int i = 0; i < PARTS_PER_N / 4; ++i) {  // 128 K=4 steps
    v2f a;
    a.x = row[4 * i + k0];
    a.y = row[4 * i + k0 + 1];
    // D = A(16x4) * ones(4x16) + C  => D[m][*] += sum_k A[m][k]
    acc = __builtin_amdgcn_wmma_f32_16x16x4_f32(
        /*neg_a=*/false, a, /*neg_b=*/false, ones,
        /*c_mod=*/(short)0, acc, /*reuse_a=*/false, /*reuse_b=*/false);
  }

  // sum of bias over channels (32 values; tiny, done redundantly per lane)
  float bs = 0.0f;
#pragma unroll
  for (int j = 0; j < C_B; ++j) bs += bias[j];

  // out[n] = rowsum / (DIVISOR * PD*PH*PW) + bias_sum = rowsum/32768 + bs
  const float scale = 1.0f / 32768.0f;
  if (lane == 0) {
#pragma unroll
    for (int v = 0; v < 8; ++v) out[v] = acc[v] * scale + bs;
  } else if (lane == 16) {
#pragma unroll
    for (int v = 0; v < 8; ++v) out[v + 8] = acc[v] * scale + bs;
  }
}

extern "C" void kernel_launch(void* const* d_in, const int* in_sizes, int n_in,
                              void* d_out, int out_size, void* d_ws,
                              size_t ws_size, hipStream_t stream) {
  (void)in_sizes; (void)n_in; (void)out_size; (void)ws_size;
  const float* o    = (const float*)d_in[0];
  const float* bias = (const float*)d_in[1];
  float* out        = (float*)d_out;
  float* partials   = (float*)d_ws;  // 8192 floats = 32 KB of scratch

  pool_partial_kernel<<<NSLABS, 256, 0, stream>>>(o, partials);
  finalize_wmma_kernel<<<1, 32, 0, stream>>>(partials, bias, out);
}